// DetectionPostprocess_6700148982203
// MI455X (gfx1250) — compile-verified
//
#include <hip/hip_runtime.h>
#include <math.h>

// ---------------- problem constants (match reference) ----------------
#define DD 64
#define HH 64
#define WW 64
#define N_ (DD * HH * WW)       // 262144 anchors per batch
#define TOPK 60
#define NMS_TOPK 20
#define THRESH 0.15f
#define NMS_THR 0.05f
#define STRIDE_ 2.0f            // 128/64 in every dim

#define NTHREADS 1024           // 32 wave32 waves per workgroup
#define TILE 4096               // floats per staged tile (4 floats/thread, b128)
#define NTILES (N_ / TILE)      // 64 tiles per batch
#define CAPA 12288              // pivot-bin candidate capacity (96KB of CDNA5's 320KB LDS)
#define CAPB 256                // capacities for later radix rounds

// order-preserving float -> uint key (larger float => larger key)
__device__ __forceinline__ unsigned f2key(float f) {
    unsigned b = __float_as_uint(f);
    return (b & 0x80000000u) ? ~b : (b | 0x80000000u);
}
__device__ __forceinline__ float key2f(unsigned k) {
    unsigned b = (k & 0x80000000u) ? (k ^ 0x80000000u) : ~k;
    return __uint_as_float(b);
}

// Staging tiles (phase 1) and the big candidate buffer (phases 3+) are
// time-disjoint, so overlay them to stay within LDS budget.
union StageCand {
    float stage[2][TILE];                                  // 32 KB
    struct { unsigned key[CAPA]; unsigned idx[CAPA]; } a;  // 96 KB
};

__global__ __launch_bounds__(NTHREADS)
void detpost_kernel(const float* __restrict__ Cls,
                    const float* __restrict__ Shape,
                    const float* __restrict__ Offset,
                    float* __restrict__ out) {
    __shared__ StageCand U;
    __shared__ unsigned candKeyB[CAPB], candIdxB[CAPB];
    __shared__ unsigned candKeyC[CAPB], candIdxC[CAPB];
    __shared__ unsigned hist[256];
    __shared__ unsigned defKey[64], defIdx[64];
    __shared__ unsigned long long srt[128];
    __shared__ unsigned sA, sPivot, sN, sDefCnt, sDstCnt;
    __shared__ float bx[TOPK * 6];
    __shared__ float ssc[TOPK];
    __shared__ int   ssup[TOPK];
    __shared__ int   skept[TOPK];
    __shared__ int   sinv[TOPK];
    __shared__ int   s_nk, s_take, s_nktot;

    const int t = threadIdx.x;
    const unsigned lane = (unsigned)(t & 31);
    const int b = blockIdx.x;
    const float* cls = Cls + (size_t)b * N_;

    if (t < 256) hist[t] = 0;
    if (t == 0) { sDefCnt = 0; sDstCnt = 0; s_nk = 0; }
    __syncthreads();

    // ---- Phase 1: 8-bin histogram of key[31:29] via wave32 ballots,
    //      double-buffered async global->LDS staging (no per-element atomics) ----
    {
        unsigned lds0 = (unsigned)(size_t)(&U.stage[0][t * 4]);
        unsigned lds1 = (unsigned)(size_t)(&U.stage[1][t * 4]);
        unsigned wcnt[8] = {0, 0, 0, 0, 0, 0, 0, 0};  // uniform across the wave
        asm volatile("global_load_async_to_lds_b128 %0, %1, off"
                     :: "v"(lds0), "v"(cls + t * 4) : "memory");
        for (int it = 0; it < NTILES; ++it) {
            int buf = it & 1;
            if (it + 1 < NTILES) {
                asm volatile("global_load_async_to_lds_b128 %0, %1, off"
                             :: "v"(buf ? lds0 : lds1),
                                "v"(cls + (size_t)(it + 1) * TILE + t * 4)
                             : "memory");
                asm volatile("s_wait_asynccnt 1" ::: "memory");
            } else {
                asm volatile("s_wait_asynccnt 0" ::: "memory");
            }
            float4 v = *(const float4*)&U.stage[buf][t * 4];
            unsigned b3[4] = { f2key(v.x) >> 29, f2key(v.y) >> 29,
                               f2key(v.z) >> 29, f2key(v.w) >> 29 };
#pragma unroll
            for (int j = 0; j < 4; ++j) {
#pragma unroll
                for (int r = 0; r < 8; ++r)
                    wcnt[r] += (unsigned)__popcll(__ballot(b3[j] == (unsigned)r));
            }
        }
        if (lane == 0) {
#pragma unroll
            for (int r = 0; r < 8; ++r)
                if (wcnt[r]) atomicAdd(&hist[r], wcnt[r]);
        }
    }
    __syncthreads();

    // ---- Phase 2: pick 3-bit pivot bin containing overall rank TOPK ----
    if (t == 0) {
        unsigned acc = 0; int pv = 0;
        for (int bb = 7; bb >= 0; --bb) {
            if (acc + hist[bb] >= (unsigned)TOPK) { pv = bb; break; }
            acc += hist[bb];
        }
        sPivot = (unsigned)pv; sA = acc; sDstCnt = 0;
    }
    __syncthreads();

    // ---- Phase 3: rescan (L2-resident), compact with wave-aggregated appends ----
    {
        unsigned pv = sPivot;
        const float4* p4 = (const float4*)cls;
        for (int it = 0; it < NTILES; ++it) {
            if (it + 4 < NTILES)
                __builtin_prefetch(&p4[(size_t)(it + 4) * (TILE / 4) + t], 0, 0);
            float4 v = p4[(size_t)it * (TILE / 4) + t];
            unsigned base = (unsigned)((it * (TILE / 4) + t) * 4);
            unsigned kk[4] = { f2key(v.x), f2key(v.y), f2key(v.z), f2key(v.w) };
#pragma unroll
            for (int j = 0; j < 4; ++j) {
                unsigned k  = kk[j];
                unsigned b3 = k >> 29;
                if (b3 > pv) {   // rare: strictly-above definites (< TOPK total)
                    unsigned p = atomicAdd(&sDefCnt, 1u);
                    if (p < 64u) { defKey[p] = k; defIdx[p] = base + j; }
                }
                bool isCand = (b3 == pv);
                unsigned long long m = __ballot(isCand);
                unsigned cnt = (unsigned)__popcll(m);
                if (cnt) {
                    unsigned basep = 0;
                    if (lane == 0) basep = atomicAdd(&sDstCnt, cnt);
                    basep = (unsigned)__shfl((int)basep, 0, 32);
                    if (isCand) {
                        unsigned pos = basep +
                            (unsigned)__popcll(m & ((1ull << lane) - 1ull));
                        if (pos < (unsigned)CAPA) {
                            U.a.key[pos] = k; U.a.idx[pos] = base + j;
                        }
                    }
                }
            }
        }
    }
    __syncthreads();
    if (t == 0) sN = (sDstCnt < (unsigned)CAPA) ? sDstCnt : (unsigned)CAPA;
    __syncthreads();

    // ---- Phase 4: four radix-select rounds in LDS over remaining 29 key bits ----
    {
        unsigned *sK = U.a.key, *sI = U.a.idx;
        unsigned *dK = candKeyB, *dI = candIdxB;
        unsigned *eK = candKeyC, *eI = candIdxC;
        const int      shifts[4] = {21, 13, 5, 0};
        const unsigned masksv[4] = {255u, 255u, 255u, 31u};
        for (int r = 0; r < 4; ++r) {
            int shift = shifts[r];
            unsigned msk = masksv[r];
            if (t < 256) hist[t] = 0;
            __syncthreads();
            unsigned n = sN;
            for (unsigned i = t; i < n; i += NTHREADS)
                atomicAdd(&hist[(sK[i] >> shift) & msk], 1u);
            __syncthreads();
            if (t == 0) {
                unsigned acc = sA; int pv = 0;
                for (int bb = (int)msk; bb >= 0; --bb) {
                    if (acc + hist[bb] >= (unsigned)TOPK) { pv = bb; break; }
                    acc += hist[bb];
                }
                sPivot = (unsigned)pv; sA = acc; sDstCnt = 0;
            }
            __syncthreads();
            unsigned pv = sPivot;
            for (unsigned i = t; i < n; i += NTHREADS) {
                unsigned by = (sK[i] >> shift) & msk;
                if (by > pv) {
                    unsigned p = atomicAdd(&sDefCnt, 1u);
                    if (p < 64u) { defKey[p] = sK[i]; defIdx[p] = sI[i]; }
                } else if (by == pv) {
                    unsigned p = atomicAdd(&sDstCnt, 1u);
                    if (p < (unsigned)CAPB) { dK[p] = sK[i]; dI[p] = sI[i]; }
                }
            }
            __syncthreads();
            if (t == 0) sN = (sDstCnt < (unsigned)CAPB) ? sDstCnt : (unsigned)CAPB;
            __syncthreads();
            unsigned *tK = sK, *tI = sI;
            sK = dK; sI = dI; dK = eK; dI = eI; eK = tK; eI = tI;
        }

        // ---- merge definites + threshold ties, bitonic sort 128 desc ----
        // sort key packs (score_key desc, index asc):  (key<<32) | ~idx
        if (t < 128) {
            unsigned long long v = 0ull;
            unsigned dc = sDefCnt; if (dc > 64u) dc = 64u;
            if ((unsigned)t < dc) {
                v = ((unsigned long long)defKey[t] << 32) | (unsigned)(~defIdx[t]);
            } else {
                unsigned j = (unsigned)t - dc;
                if (j < sN)
                    v = ((unsigned long long)sK[j] << 32) | (unsigned)(~sI[j]);
            }
            srt[t] = v;
        }
        __syncthreads();
        for (int k = 2; k <= 128; k <<= 1) {
            for (int j = k >> 1; j > 0; j >>= 1) {
                if (t < 128) {
                    int ixj = t ^ j;
                    if (ixj > t) {
                        unsigned long long a = srt[t], c = srt[ixj];
                        bool descBlock = ((t & k) == 0);
                        if (descBlock ? (a < c) : (a > c)) { srt[t] = c; srt[ixj] = a; }
                    }
                }
                __syncthreads();
            }
        }
    }

    // ---- Phase 5: gather Shape/Offset for the 60 winners, build boxes ----
    if (t < TOPK) {
        unsigned long long v = srt[t];
        unsigned key = (unsigned)(v >> 32);
        unsigned idx = ~(unsigned)(v & 0xffffffffu);
        float f  = key2f(key);
        float sc = 1.0f / (1.0f + expf(-f));
        ssc[t]   = sc;
        int valid = (sc > THRESH) && (idx < (unsigned)N_);
        ssup[t]  = !valid;
        skept[t] = 0;
        unsigned gi = (idx < (unsigned)N_) ? idx : 0u;
        float az = (float)(gi >> 12);
        float ay = (float)((gi >> 6) & 63u);
        float ax = (float)(gi & 63u);
        const size_t sb = (size_t)b * 3 * N_;
        float oz = Offset[sb + 0 * N_ + gi];
        float oy = Offset[sb + 1 * N_ + gi];
        float ox = Offset[sb + 2 * N_ + gi];
        float hz = Shape[sb + 0 * N_ + gi];
        float hy = Shape[sb + 1 * N_ + gi];
        float hx = Shape[sb + 2 * N_ + gi];
        bx[t * 6 + 0] = (az + oz) * STRIDE_;
        bx[t * 6 + 1] = (ay + oy) * STRIDE_;
        bx[t * 6 + 2] = (ax + ox) * STRIDE_;
        bx[t * 6 + 3] = 2.0f * hz;
        bx[t * 6 + 4] = 2.0f * hy;
        bx[t * 6 + 5] = 2.0f * hx;
    }
    __syncthreads();

    // ---- Phase 6: greedy NMS (serial over i, parallel over j) ----
    float jlo0 = 0.f, jlo1 = 0.f, jlo2 = 0.f;
    float jhi0 = 0.f, jhi1 = 0.f, jhi2 = 0.f, jvol = 0.f;
    if (t < TOPK) {
        float c0 = bx[t*6+0], c1 = bx[t*6+1], c2 = bx[t*6+2];
        float s0 = bx[t*6+3], s1 = bx[t*6+4], s2 = bx[t*6+5];
        jlo0 = c0 - 0.5f*s0; jhi0 = c0 + 0.5f*s0;
        jlo1 = c1 - 0.5f*s1; jhi1 = c1 + 0.5f*s1;
        jlo2 = c2 - 0.5f*s2; jhi2 = c2 + 0.5f*s2;
        jvol = s0 * s1 * s2;
    }
    for (int i = 0; i < TOPK; ++i) {
        if (t == 0) {
            int take = (!ssup[i]) && (s_nk < NMS_TOPK);
            s_take = take;
            if (take) { skept[i] = 1; s_nk++; }
        }
        __syncthreads();
        if (s_take && t < TOPK) {
            float c0 = bx[i*6+0], c1 = bx[i*6+1], c2 = bx[i*6+2];
            float s0 = bx[i*6+3], s1 = bx[i*6+4], s2 = bx[i*6+5];
            float d0 = fmaxf(fminf(c0 + 0.5f*s0, jhi0) - fmaxf(c0 - 0.5f*s0, jlo0), 0.f);
            float d1 = fmaxf(fminf(c1 + 0.5f*s1, jhi1) - fmaxf(c1 - 0.5f*s1, jlo1), 0.f);
            float d2 = fmaxf(fminf(c2 + 0.5f*s2, jhi2) - fmaxf(c2 - 0.5f*s2, jlo2), 0.f);
            float inter = d0 * d1 * d2;
            float uni   = s0 * s1 * s2 + jvol - inter;
            float iou   = inter / fmaxf(uni, 1e-8f);
            if (iou > NMS_THR) ssup[t] = 1;
        }
        __syncthreads();
    }

    // ---- Phase 7: compact kept rows; fill the rest with -1 ----
    if (t == 0) {
        int pos = 0;
        for (int i = 0; i < TOPK; ++i) if (skept[i]) sinv[pos++] = i;
        s_nktot = pos;
    }
    __syncthreads();
    if (t < TOPK) {
        float* row = out + ((size_t)b * TOPK + t) * 8;
        if (t < s_nktot) {
            int i = sinv[t];
            row[0] = 1.0f;      row[1] = ssc[i];
            row[2] = bx[i*6+0]; row[3] = bx[i*6+1]; row[4] = bx[i*6+2];
            row[5] = bx[i*6+3]; row[6] = bx[i*6+4]; row[7] = bx[i*6+5];
        } else {
#pragma unroll
            for (int c = 0; c < 8; ++c) row[c] = -1.0f;
        }
    }
}

extern "C" void kernel_launch(void* const* d_in, const int* in_sizes, int n_in,
                              void* d_out, int out_size, void* d_ws, size_t ws_size,
                              hipStream_t stream) {
    (void)n_in; (void)out_size; (void)d_ws; (void)ws_size;
    const float* Cls    = (const float*)d_in[0];
    const float* Shape  = (const float*)d_in[1];
    const float* Offset = (const float*)d_in[2];
    float* out = (float*)d_out;
    int B = in_sizes[0] / N_;   // 32
    detpost_kernel<<<dim3(B), dim3(NTHREADS), 0, stream>>>(Cls, Shape, Offset, out);
}